// NNFOwithBayesianJumps_73839077753176
// MI455X (gfx1250) — compile-verified
//
#include <hip/hip_runtime.h>
#include <hip/hip_bf16.h>
#include <math.h>

// Problem constants (match reference)
#define B_    8192
#define D_    32
#define H_    128
#define P2_   64     // 2*D (p width, also PH)
#define PREP_ 10
#define GIN_  320    // D*PREP
#define G3_   384    // 3*H
#define CS_   8
#define CH_   32
#define NT_   16
#define NO_   4096
#define DT_   0.1f
#define NPOST_ 4
#define LOGC_  0.91893853320467274178f  // log(sqrt(2*pi))

typedef __attribute__((ext_vector_type(2))) float v2f;
typedef __attribute__((ext_vector_type(8))) float v8f;

__device__ __forceinline__ float sigmoidf_(float x) { return 1.0f / (1.0f + expf(-x)); }

// ---- WMMA f32 16x16x4 helpers -------------------------------------------
// A (MxK row-major, stride lda): lane holds {A[m][k0], A[m][k0+1]},
//   m = m0 + (lane&15), k0 = k + ((lane>>4)<<1)   (ISA 7.12.2, 32-bit A 16x4)
// B (KxN row-major, stride ldb): lane holds {B[k0][n], B[k0+1][n]},
//   n = n0 + (lane&15)  (mirror of A layout)
// C/D: reg j -> row m0 + j + ((lane>>4)<<3), col n0 + (lane&15)

__device__ __forceinline__ v2f load_afrag(const float* A, int lda, int m0, int k, int lane) {
    int m  = m0 + (lane & 15);
    int kk = k + ((lane >> 4) << 1);
    v2f a;
    a.x = A[m * lda + kk];
    a.y = A[m * lda + kk + 1];
    return a;
}

__device__ __forceinline__ v2f load_bfrag(const float* Bm, int ldb, int k, int n0, int lane) {
    int n  = n0 + (lane & 15);
    int kk = k + ((lane >> 4) << 1);
    v2f b;
    b.x = Bm[kk * ldb + n];
    b.y = Bm[(kk + 1) * ldb + n];
    return b;
}

__device__ __forceinline__ v8f wmma4(v2f a, v2f b, v8f c) {
    // (neg_a, A, neg_b, B, c_mod, C, reuse_a, reuse_b)
    return __builtin_amdgcn_wmma_f32_16x16x4_f32(false, a, false, b, (short)0, c, false, false);
}

// ---- init: h = tanh(relu(cov@cm_w1+b1)@cm_w2+b2); also zero loss ---------
__global__ void k_init(const float* __restrict__ cov,
                       const float* __restrict__ w1, const float* __restrict__ b1,
                       const float* __restrict__ w2, const float* __restrict__ b2,
                       float* __restrict__ h, float* __restrict__ loss) {
    __shared__ float t_s[16][CH_];
    const int tid = threadIdx.x;           // 128 threads
    const int m0  = blockIdx.x * 16;
    if (blockIdx.x == 0 && tid == 0) *loss = 0.0f;
    for (int i = tid; i < 16 * CH_; i += 128) {
        int r = i / CH_, c = i % CH_;
        float acc = b1[c];
        #pragma unroll
        for (int k = 0; k < CS_; ++k) acc += cov[(m0 + r) * CS_ + k] * w1[k * CH_ + c];
        t_s[r][c] = fmaxf(acc, 0.0f);
    }
    __syncthreads();
    for (int i = tid; i < 16 * H_; i += 128) {
        int r = i / H_, c = i % H_;
        float acc = b2[c];
        #pragma unroll
        for (int k = 0; k < CH_; ++k) acc += t_s[r][k] * w2[k * H_ + c];
        h[(m0 + r) * H_ + c] = tanhf(acc);
    }
}

// ---- p = relu(h@p_w1+b1)@p_w2+b2 ;  4 waves, one 16x16 N-tile each -------
__global__ void k_pmodel(const float* __restrict__ h,
                         const float* __restrict__ w1, const float* __restrict__ b1,
                         const float* __restrict__ w2, const float* __restrict__ b2,
                         float* __restrict__ p) {
    __shared__ float t_s[16][P2_];         // relu intermediate [16 x 64]
    const int tid = threadIdx.x, lane = tid & 31, wv = tid >> 5;  // 128 threads
    const int m0 = blockIdx.x * 16, n0 = wv * 16;

    v8f acc = {};
    for (int k = 0; k < H_; k += 4)
        acc = wmma4(load_afrag(h, H_, m0, k, lane), load_bfrag(w1, P2_, k, n0, lane), acc);
    const int col = n0 + (lane & 15), rb = (lane >> 4) << 3;
    const float bv = b1[col];
    #pragma unroll
    for (int j = 0; j < 8; ++j) t_s[rb + j][col] = fmaxf(acc[j] + bv, 0.0f);
    __syncthreads();

    v8f acc2 = {};
    for (int k = 0; k < P2_; k += 4)
        acc2 = wmma4(load_afrag(&t_s[0][0], P2_, 0, k, lane), load_bfrag(w2, P2_, k, n0, lane), acc2);
    const float bv2 = b2[col];
    #pragma unroll
    for (int j = 0; j < 8; ++j) p[(m0 + rb + j) * P2_ + col] = acc2[j] + bv2;
}

// ---- h += DT * gru_ode(p, h) ; 8 waves, one N-tile each ------------------
__global__ void k_ode(float* __restrict__ h, const float* __restrict__ p,
                      const float* __restrict__ xz_w, const float* __restrict__ xz_b,
                      const float* __restrict__ hz_w,
                      const float* __restrict__ xn_w, const float* __restrict__ xn_b,
                      const float* __restrict__ hn_w) {
    __shared__ float zh_s[16][H_];
    const int tid = threadIdx.x, lane = tid & 31, wv = tid >> 5;  // 256 threads
    const int m0 = blockIdx.x * 16, n0 = wv * 16;

    v8f az = {};
    for (int k = 0; k < P2_; k += 4)
        az = wmma4(load_afrag(p, P2_, m0, k, lane), load_bfrag(xz_w, H_, k, n0, lane), az);
    for (int k = 0; k < H_; k += 4)
        az = wmma4(load_afrag(h, H_, m0, k, lane), load_bfrag(hz_w, H_, k, n0, lane), az);

    const int col = n0 + (lane & 15), rb = (lane >> 4) << 3;
    const float bz = xz_b[col];
    float z[8];
    #pragma unroll
    for (int j = 0; j < 8; ++j) {
        z[j] = sigmoidf_(az[j] + bz);
        zh_s[rb + j][col] = z[j] * h[(m0 + rb + j) * H_ + col];
    }
    __syncthreads();

    v8f an = {};
    for (int k = 0; k < P2_; k += 4)
        an = wmma4(load_afrag(p, P2_, m0, k, lane), load_bfrag(xn_w, H_, k, n0, lane), an);
    for (int k = 0; k < H_; k += 4)
        an = wmma4(load_afrag(&zh_s[0][0], H_, 0, k, lane), load_bfrag(hn_w, H_, k, n0, lane), an);

    const float bn = xn_b[col];
    #pragma unroll
    for (int j = 0; j < 8; ++j) {
        float n = tanhf(an[j] + bn);
        int row = m0 + rb + j;
        float hv = h[row * H_ + col];
        h[row * H_ + col] = hv + DT_ * (1.0f - z[j]) * (n - hv);
    }
}

// ---- Bayesian jump: gather, loss, prep, GRUCell via WMMA, scatter --------
__global__ void k_jump(int t,
                       const float* __restrict__ X, const float* __restrict__ M,
                       const int* __restrict__ obs_idx,
                       float* __restrict__ h, const float* __restrict__ p,
                       const float* __restrict__ w_ih, const float* __restrict__ w_hh,
                       const float* __restrict__ b_ih, const float* __restrict__ b_hh,
                       const float* __restrict__ w_prep, const float* __restrict__ bias_prep,
                       float* __restrict__ loss_part) {
    __shared__ int   idx_s[16];
    __shared__ float hold_s[16][H_];
    __shared__ float gin_s[16][GIN_];
    __shared__ float gi_s[16][G3_];
    __shared__ float gh_s[16][G3_];
    __shared__ float red[256];
    const int tid = threadIdx.x, lane = tid & 31, wv = tid >> 5;  // 256 threads
    const int o0 = blockIdx.x * 16;
    const float* Xt = X + (size_t)t * NO_ * D_;
    const float* Mt = M + (size_t)t * NO_ * D_;

    if (tid < 16) idx_s[tid] = obs_idx[t * NO_ + o0 + tid];
    __syncthreads();

    for (int i = tid; i < 16 * H_; i += 256) {
        int o = i >> 7, j = i & 127;
        hold_s[o][j] = h[idx_s[o] * H_ + j];
    }

    float lsum = 0.0f;
    for (int i = tid; i < 16 * D_; i += 256) {     // loss terms, per (o,d)
        int o = i / D_, d = i % D_;
        int gb = o0 + o, b = idx_s[o];
        float mean = p[b * P2_ + d];
        float lv   = p[b * P2_ + D_ + d];
        float x    = Xt[gb * D_ + d];
        float m    = Mt[gb * D_ + d];
        float err  = (x - mean) * expf(-0.5f * lv);
        lsum += 0.5f * ((err * err + lv + 2.0f * LOGC_) * m);
    }

    for (int i = tid; i < 16 * GIN_; i += 256) {   // gin = relu(feats . w_prep + b) * M
        int o = i / GIN_, c = i % GIN_;
        int d = c / PREP_, pi = c - d * PREP_;
        int gb = o0 + o, b = idx_s[o];
        float mean = p[b * P2_ + d];
        float lv   = p[b * P2_ + D_ + d];
        float x    = Xt[gb * D_ + d];
        float m    = Mt[gb * D_ + d];
        float err  = (x - mean) * expf(-0.5f * lv);
        const float* wp = w_prep + (size_t)(d * 4) * PREP_ + pi;
        float a = x * wp[0] + mean * wp[PREP_] + lv * wp[2 * PREP_] + err * wp[3 * PREP_]
                + bias_prep[d * PREP_ + pi];
        gin_s[o][c] = fmaxf(a, 0.0f) * m;
    }
    __syncthreads();

    // gi = gin @ w_ih + b_ih ; gh = h_g @ w_hh + b_hh   (24 tiles: 8 waves x 3)
    for (int s = 0; s < 3; ++s) {
        const int n0 = (wv * 3 + s) * 16;
        v8f gi = {};
        for (int k = 0; k < GIN_; k += 4)
            gi = wmma4(load_afrag(&gin_s[0][0], GIN_, 0, k, lane), load_bfrag(w_ih, G3_, k, n0, lane), gi);
        v8f gh = {};
        for (int k = 0; k < H_; k += 4)
            gh = wmma4(load_afrag(&hold_s[0][0], H_, 0, k, lane), load_bfrag(w_hh, G3_, k, n0, lane), gh);
        const int col = n0 + (lane & 15), rb = (lane >> 4) << 3;
        const float bi = b_ih[col], bh = b_hh[col];
        #pragma unroll
        for (int j = 0; j < 8; ++j) {
            gi_s[rb + j][col] = gi[j] + bi;
            gh_s[rb + j][col] = gh[j] + bh;
        }
    }
    __syncthreads();

    for (int i = tid; i < 16 * H_; i += 256) {     // gates + scatter
        int o = i >> 7, j = i & 127;
        float r = sigmoidf_(gi_s[o][j] + gh_s[o][j]);
        float z = sigmoidf_(gi_s[o][H_ + j] + gh_s[o][H_ + j]);
        float n = tanhf(gi_s[o][2 * H_ + j] + r * gh_s[o][2 * H_ + j]);
        float hv = hold_s[o][j];
        h[idx_s[o] * H_ + j] = (1.0f - z) * n + z * hv;
    }

    // deterministic per-block loss partial
    red[tid] = lsum;
    __syncthreads();
    for (int s2 = 128; s2 > 0; s2 >>= 1) {
        if (tid < s2) red[tid] += red[tid + s2];
        __syncthreads();
    }
    if (tid == 0) loss_part[t * (NO_ / 16) + blockIdx.x] = red[0];
}

// ---- deterministic final loss reduction ----------------------------------
__global__ void k_loss_final(const float* __restrict__ parts, float* __restrict__ loss) {
    __shared__ float red[256];
    const int tid = threadIdx.x;
    float s = 0.0f;
    for (int i = tid; i < NT_ * (NO_ / 16); i += 256) s += parts[i];
    red[tid] = s;
    __syncthreads();
    for (int k = 128; k > 0; k >>= 1) {
        if (tid < k) red[tid] += red[tid + k];
        __syncthreads();
    }
    if (tid == 0) *loss = red[0];
}

extern "C" void kernel_launch(void* const* d_in, const int* in_sizes, int n_in,
                              void* d_out, int out_size, void* d_ws, size_t ws_size,
                              hipStream_t stream) {
    const float* X       = (const float*)d_in[0];
    const float* M       = (const float*)d_in[1];
    const int*   obs_idx = (const int*)d_in[2];
    const float* cov     = (const float*)d_in[3];
    const float* cm_w1   = (const float*)d_in[4];
    const float* cm_b1   = (const float*)d_in[5];
    const float* cm_w2   = (const float*)d_in[6];
    const float* cm_b2   = (const float*)d_in[7];
    const float* p_w1    = (const float*)d_in[8];
    const float* p_b1    = (const float*)d_in[9];
    const float* p_w2    = (const float*)d_in[10];
    const float* p_b2    = (const float*)d_in[11];
    const float* xz_w    = (const float*)d_in[12];
    const float* xz_b    = (const float*)d_in[13];
    const float* hz_w    = (const float*)d_in[14];
    const float* xn_w    = (const float*)d_in[15];
    const float* xn_b    = (const float*)d_in[16];
    const float* hn_w    = (const float*)d_in[17];
    const float* w_ih    = (const float*)d_in[18];
    const float* w_hh    = (const float*)d_in[19];
    const float* b_ih    = (const float*)d_in[20];
    const float* b_hh    = (const float*)d_in[21];
    const float* w_prep  = (const float*)d_in[22];
    const float* b_prep  = (const float*)d_in[23];

    float* h    = (float*)d_out;                       // [B,H]
    float* p    = h + (size_t)B_ * H_;                 // [B,2D]
    float* loss = p + (size_t)B_ * P2_;                // [1]
    float* loss_part = (float*)d_ws;                   // [NT * NO/16]

    const int MB = B_ / 16;   // 512 row-tiles
    const int OB = NO_ / 16;  // 256 obs-tiles

    k_init<<<MB, 128, 0, stream>>>(cov, cm_w1, cm_b1, cm_w2, cm_b2, h, loss);
    k_pmodel<<<MB, 128, 0, stream>>>(h, p_w1, p_b1, p_w2, p_b2, p);

    for (int t = 0; t < NT_; ++t) {
        k_ode<<<MB, 256, 0, stream>>>(h, p, xz_w, xz_b, hz_w, xn_w, xn_b, hn_w);
        k_pmodel<<<MB, 128, 0, stream>>>(h, p_w1, p_b1, p_w2, p_b2, p);
        k_jump<<<OB, 256, 0, stream>>>(t, X, M, obs_idx, h, p,
                                       w_ih, w_hh, b_ih, b_hh, w_prep, b_prep, loss_part);
        k_pmodel<<<MB, 128, 0, stream>>>(h, p_w1, p_b1, p_w2, p_b2, p);
    }
    for (int s = 0; s < NPOST_; ++s) {
        k_ode<<<MB, 256, 0, stream>>>(h, p, xz_w, xz_b, hz_w, xn_w, xn_b, hn_w);
        k_pmodel<<<MB, 128, 0, stream>>>(h, p_w1, p_b1, p_w2, p_b2, p);
    }
    k_loss_final<<<1, 256, 0, stream>>>(loss_part, loss);
}